// GCN_12206297055601
// MI455X (gfx1250) — compile-verified
//
#include <hip/hip_runtime.h>

// ---- problem sizes (fixed by the reference) ----
#define NNODES 100000
#define NEDGES 3200000
#define NFEATS 256
#define NHID   128
#define NCLS   16
#define TEXTN  50000

typedef __attribute__((ext_vector_type(2))) float v2f;
typedef __attribute__((ext_vector_type(8))) float v8f;

// -------------------------------------------------------------------------
// Elementwise helpers
// -------------------------------------------------------------------------
__global__ void zero_f32(float* __restrict__ p, long n) {
  long i = (long)blockIdx.x * blockDim.x + threadIdx.x;
  long stride = (long)gridDim.x * blockDim.x;
  for (; i < n; i += stride) p[i] = 0.0f;
}

__global__ void bias_relu_f32(float* __restrict__ h, const float* __restrict__ b,
                              long n, int mask) {
  long i = (long)blockIdx.x * blockDim.x + threadIdx.x;
  long stride = (long)gridDim.x * blockDim.x;
  for (; i < n; i += stride) {
    float v = h[i] + b[(int)(i & mask)];
    h[i] = v > 0.0f ? v : 0.0f;
  }
}

__global__ void bias_add_f32(float* __restrict__ o, const float* __restrict__ b,
                             long n, int mask) {
  long i = (long)blockIdx.x * blockDim.x + threadIdx.x;
  long stride = (long)gridDim.x * blockDim.x;
  for (; i < n; i += stride) o[i] += b[(int)(i & mask)];
}

// -------------------------------------------------------------------------
// Macro-tiled fp32 GEMM via V_WMMA_F32_16X16X4_F32.
// One wave owns a 32x64 C macro-tile: 2 M-tiles x 4 N-tiles = 8 accumulators.
// Per K-step (K += 4): 2 x b64 A loads + 8 x b32 B loads -> 8 WMMAs.
// K, Nc are template constants so all strides become immediate offsets.
// Requires M % 32 == 0 and Nc % 64 == 0 (true for gemm1/gemm2 here).
// -------------------------------------------------------------------------
template <int K, int Nc>
__global__ void gemm_wmma_2x4(const float* __restrict__ A,
                              const float* __restrict__ B,
                              float* __restrict__ C,
                              int M) {
  const int ntN   = Nc / 64;                 // macro N tiles
  const int tiles = (M / 32) * ntN;
  const int wave  = blockIdx.x * (blockDim.x >> 5) + ((int)threadIdx.x >> 5);
  if (wave >= tiles) return;                 // wave-uniform

  const int tm   = wave / ntN;
  const int tn   = wave - tm * ntN;
  const int lane = (int)threadIdx.x & 31;
  const int half = lane >> 4;                // 0: K={0,1}, 1: K={2,3}
  const int l16  = lane & 15;

  const int m0 = tm * 32;
  const int n0 = tn * 64;

  const float* arow0 = A + (size_t)(m0 + l16) * K;        // M-tile 0
  const float* arow1 = A + (size_t)(m0 + 16 + l16) * K;   // M-tile 1
  const int col0 = n0 + l16;                               // N-tile j at col0 + 16*j

  v8f acc[2][4] = {};

#pragma unroll 2
  for (int k = 0; k < K; k += 4) {
    const int kk = k + half * 2;
    v2f a0 = *(const v2f*)(arow0 + kk);
    v2f a1 = *(const v2f*)(arow1 + kk);
    const float* brow0 = B + (size_t)kk * Nc + col0;
    const float* brow1 = brow0 + Nc;
    v2f b[4];
#pragma unroll
    for (int j = 0; j < 4; ++j) {
      b[j].x = brow0[j * 16];
      b[j].y = brow1[j * 16];
    }
#pragma unroll
    for (int j = 0; j < 4; ++j) {
      acc[0][j] = __builtin_amdgcn_wmma_f32_16x16x4_f32(false, a0, false, b[j],
                                                        (short)0, acc[0][j], false, false);
      acc[1][j] = __builtin_amdgcn_wmma_f32_16x16x4_f32(false, a1, false, b[j],
                                                        (short)0, acc[1][j], false, false);
    }
  }

#pragma unroll
  for (int i = 0; i < 2; ++i) {
    const int r0 = m0 + i * 16 + half * 8;
#pragma unroll
    for (int j = 0; j < 4; ++j) {
      float* cp = C + (size_t)r0 * Nc + col0 + j * 16;
#pragma unroll
      for (int v = 0; v < 8; ++v) cp[(size_t)v * Nc] = acc[i][j][v];
    }
  }
}

// -------------------------------------------------------------------------
// Simple 16x16-per-wave fp32 WMMA GEMM with fused bias (classifier heads).
// Handles any M % 16 == 0, Nc % 16 == 0.
// -------------------------------------------------------------------------
template <int K, int Nc>
__global__ void gemm_wmma_bias(const float* __restrict__ A,
                               const float* __restrict__ B,
                               const float* __restrict__ bias,
                               float* __restrict__ C,
                               int M) {
  const int ntN   = Nc >> 4;
  const int tiles = (M >> 4) * ntN;
  const int wave  = blockIdx.x * (blockDim.x >> 5) + ((int)threadIdx.x >> 5);
  if (wave >= tiles) return;                 // wave-uniform

  const int mt   = wave / ntN;
  const int nt   = wave - mt * ntN;
  const int lane = (int)threadIdx.x & 31;
  const int half = lane >> 4;
  const int l16  = lane & 15;

  const float* arow = A + (size_t)(mt * 16 + l16) * K;
  const int col = nt * 16 + l16;

  v8f acc = {};
#pragma unroll 4
  for (int k = 0; k < K; k += 4) {
    const int kk = k + half * 2;
    v2f a = *(const v2f*)(arow + kk);
    v2f b;
    b.x = B[(size_t)kk * Nc + col];
    b.y = B[(size_t)(kk + 1) * Nc + col];
    acc = __builtin_amdgcn_wmma_f32_16x16x4_f32(false, a, false, b,
                                                (short)0, acc, false, false);
  }

  const float bv = bias[col];
  const int r0 = mt * 16 + half * 8;
#pragma unroll
  for (int v = 0; v < 8; ++v)
    C[(size_t)(r0 + v) * Nc + col] = acc[v] + bv;
}

// -------------------------------------------------------------------------
// COO SpMM scatter: out[rows[e], :] += vals[e] * dense[cols[e], :]
// One wave per edge; lanes stride the feature dim (coalesced 128B gather,
// f32 scatter-atomics resolved in the 192MB L2).
// -------------------------------------------------------------------------
template <int D>
__global__ void spmm_scatter(const int* __restrict__ rows,
                             const int* __restrict__ cols,
                             const float* __restrict__ vals,
                             const float* __restrict__ dense,
                             float* __restrict__ out) {
  const long e = (long)blockIdx.x * (blockDim.x >> 5) + ((int)threadIdx.x >> 5);
  if (e >= (long)NEDGES) return;
  const int lane = (int)threadIdx.x & 31;

  const int   r = rows[e];
  const int   c = cols[e];
  const float v = vals[e];

  const float* __restrict__ src = dense + (size_t)c * D;
  float* __restrict__ dst = out + (size_t)r * D;

#pragma unroll
  for (int i = 0; i < D / 32; ++i) {
    const int idx = lane + i * 32;
    atomicAdd(dst + idx, v * src[idx]);
  }
}

// -------------------------------------------------------------------------
// Orchestration. d_in order: x, adj_rows, adj_cols, adj_vals,
//   W1, b1, W2, b2, cW1, cb1, cW2, cb2.
// d_out: [out (N*256) | text_cls (50000*16) | img_cls (50000*16)].
// Workspace: h at 0 (51.2MB); xW1 at 51.2MB, later overwritten by t2
//   (xW1 is dead once h is built) -> peak ws = 153.6MB.
// -------------------------------------------------------------------------
extern "C" void kernel_launch(void* const* d_in, const int* in_sizes, int n_in,
                              void* d_out, int out_size, void* d_ws, size_t ws_size,
                              hipStream_t stream) {
  const float* x    = (const float*)d_in[0];
  const int*   rows = (const int*)  d_in[1];
  const int*   cols = (const int*)  d_in[2];
  const float* vals = (const float*)d_in[3];
  const float* W1   = (const float*)d_in[4];
  const float* b1   = (const float*)d_in[5];
  const float* W2   = (const float*)d_in[6];
  const float* b2   = (const float*)d_in[7];
  const float* cW1  = (const float*)d_in[8];
  const float* cb1  = (const float*)d_in[9];
  const float* cW2  = (const float*)d_in[10];
  const float* cb2  = (const float*)d_in[11];

  float* out      = (float*)d_out;                       // [N, 256]
  float* text_cls = out + (size_t)NNODES * NFEATS;       // [50000, 16]
  float* img_cls  = text_cls + (size_t)TEXTN * NCLS;     // [50000, 16]

  float* h   = (float*)d_ws;                                           // [N,128]
  float* buf = (float*)((char*)d_ws + (size_t)NNODES * NHID * sizeof(float));
  // buf holds xW1 [N,128] first, then t2 [N,256]

  // h = 0
  zero_f32<<<2048, 256, 0, stream>>>(h, (long)NNODES * NHID);

  // xW1 = x @ W1               (100000x256 @ 256x128), 32x64 macro-tiles
  {
    const int tiles = (NNODES / 32) * (NHID / 64);       // 6250
    gemm_wmma_2x4<NFEATS, NHID><<<(tiles + 7) / 8, 256, 0, stream>>>(x, W1, buf, NNODES);
  }

  // h += scatter(adj * xW1)
  spmm_scatter<NHID><<<(NEDGES + 7) / 8, 256, 0, stream>>>(rows, cols, vals, buf, h);

  // h = relu(h + b1)
  bias_relu_f32<<<2048, 256, 0, stream>>>(h, b1, (long)NNODES * NHID, NHID - 1);

  // out = 0
  zero_f32<<<4096, 256, 0, stream>>>(out, (long)NNODES * NFEATS);

  // t2 = h @ W2                (100000x128 @ 128x256), reusing xW1 space
  {
    const int tiles = (NNODES / 32) * (NFEATS / 64);     // 12500
    gemm_wmma_2x4<NHID, NFEATS><<<(tiles + 7) / 8, 256, 0, stream>>>(h, W2, buf, NNODES);
  }

  // out += scatter(adj * t2)
  spmm_scatter<NFEATS><<<(NEDGES + 7) / 8, 256, 0, stream>>>(rows, cols, vals, buf, out);

  // out += b2
  bias_add_f32<<<4096, 256, 0, stream>>>(out, b2, (long)NNODES * NFEATS, NFEATS - 1);

  // text_cls = out[:50000] @ cW1 + cb1 ; img_cls = out[50000:] @ cW2 + cb2
  {
    const int tiles = (TEXTN / 16) * (NCLS / 16);        // 3125
    gemm_wmma_bias<NFEATS, NCLS><<<(tiles + 7) / 8, 256, 0, stream>>>(out, cW1, cb1,
                                                                      text_cls, TEXTN);
    gemm_wmma_bias<NFEATS, NCLS><<<(tiles + 7) / 8, 256, 0, stream>>>(
        out + (size_t)TEXTN * NFEATS, cW2, cb2, img_cls, TEXTN);
  }
}